// Loss_9174050144512
// MI455X (gfx1250) — compile-verified
//
#include <hip/hip_runtime.h>

// YOLO loss for MI455X (gfx1250, wave32). Memory-bound streaming reduction:
// ~193 MB of fp32 traffic -> ~8.3 us floor at 23.3 TB/s.
// Stage 1: one thread per (image, cell); coalesced strided loads; per-cell
//          loss in registers; wave reduction via V_WMMA_F32_16X16X4_F32
//          (B = ones => exact fp32 cross-lane sum); block partial -> d_ws.
// Stage 2: single block reduces 3136 partials deterministically -> d_out[0].

namespace cfg {
constexpr int   S      = 14;
constexpr int   SS     = 196;          // S*S
constexpr int   D      = 30;           // 5*B + C
constexpr int   NBATCH = 4096;
constexpr int   CELLS  = NBATCH * SS;  // 802816
constexpr int   BLOCK  = 256;          // 8 waves (wave32)
constexpr int   GRID   = CELLS / BLOCK;// 3136, exact
constexpr float LC     = 5.0f;         // LAMBDA_COORD
constexpr float LN     = 0.5f;         // LAMBDA_NOOBJ
}

typedef float v2f __attribute__((ext_vector_type(2)));
typedef float v8f __attribute__((ext_vector_type(8)));

// Exact fp32 sum of one value per lane across a wave32, using the f32 WMMA.
// A(16x4): lane L<16 -> A[L][0]=x, lane L>=16 -> A[L-16][2]=x (VGPR1 = 0).
// B(4x16) = 1.0 everywhere. D[m][n] = x[m] + x[m+16].
// Lanes 0-15 hold D rows 0-7 in c[0..7]; lanes 16-31 hold rows 8-15.
__device__ __forceinline__ float wave_sum32(float x) {
  v2f a; a[0] = x;    a[1] = 0.0f;
  v2f b; b[0] = 1.0f; b[1] = 1.0f;
  v8f c = {};
  c = __builtin_amdgcn_wmma_f32_16x16x4_f32(
      /*neg_a=*/false, a, /*neg_b=*/false, b,
      /*c_mod=*/(short)0, c, /*reuse_a=*/false, /*reuse_b=*/false);
  float s = c[0] + c[1] + c[2] + c[3] + c[4] + c[5] + c[6] + c[7];
  return __shfl(s, 0, 32) + __shfl(s, 16, 32);
}

__global__ __launch_bounds__(cfg::BLOCK)
void yolo_loss_stage1(const float* __restrict__ pred,
                      const float* __restrict__ tgt,
                      float* __restrict__ block_sums) {
  const int tid  = blockIdx.x * cfg::BLOCK + threadIdx.x;
  const int n    = tid / cfg::SS;
  const int cell = tid - n * cfg::SS;

  const float* p = pred + (size_t)n * cfg::D * cfg::SS + cell;
  const float* t = tgt  + (size_t)n * cfg::D * cfg::SS + cell;

  // Load all 30 channels of both tensors (coalesced per channel across the
  // wave: consecutive lanes -> consecutive cells -> contiguous 128B).
  float pv[cfg::D], tv[cfg::D];
#pragma unroll
  for (int d = 0; d < cfg::D; ++d) {
    pv[d] = p[d * cfg::SS];
    tv[d] = t[d * cfg::SS];
  }

  const float conf  = tv[4];                       // == tv[9]
  const float obj   = (conf > 0.0f)  ? 1.0f : 0.0f;
  const float noobj = (conf == 0.0f) ? 1.0f : 0.0f;

  // no-object confidence loss (both boxes' conf channels: 4 and 9)
  const float dc0 = pv[4] - tv[4];
  const float dc1 = pv[9] - tv[9];
  const float l_noobj = dc0 * dc0 + dc1 * dc1;

  // Target box = target's box 0 (channels 0..3), per reference.
  const float invS = 1.0f / (float)cfg::S;
  const float t_cx  = tv[0] * invS,            t_cy  = tv[1] * invS;
  const float t_ltx = t_cx - 0.5f * tv[2],     t_lty = t_cy - 0.5f * tv[3];
  const float t_rbx = t_cx + 0.5f * tv[2],     t_rby = t_cy + 0.5f * tv[3];
  const float t_area = (t_rbx - t_ltx) * (t_rby - t_lty);

  float iou0, iou1;
#pragma unroll
  for (int bb = 0; bb < 2; ++bb) {
    const int o = 5 * bb;
    const float p_cx  = pv[o + 0] * invS,        p_cy  = pv[o + 1] * invS;
    const float p_ltx = p_cx - 0.5f * pv[o + 2], p_lty = p_cy - 0.5f * pv[o + 3];
    const float p_rbx = p_cx + 0.5f * pv[o + 2], p_rby = p_cy + 0.5f * pv[o + 3];
    const float p_area = (p_rbx - p_ltx) * (p_rby - p_lty);
    const float ltx = fmaxf(t_ltx, p_ltx), lty = fmaxf(t_lty, p_lty);
    const float rbx = fminf(t_rbx, p_rbx), rby = fminf(t_rby, p_rby);
    const float wx  = fmaxf(rbx - ltx, 0.0f), wy = fmaxf(rby - lty, 0.0f);
    const float inter = wx * wy;
    const float uni   = t_area + p_area - inter;
    const float iou   = inter / ((uni == 0.0f) ? 1.0f : uni);
    if (bb == 0) iou0 = iou; else iou1 = iou;
  }

  // argmax (first max on ties, matching jnp.argmax) + stop_gradient max
  const bool  r1   = iou1 > iou0;
  const float miou = fmaxf(iou0, iou1);

  // Responsible box, selected with cndmask (no dynamic indexing -> no scratch)
  const float bpx = r1 ? pv[5] : pv[0];
  const float bpy = r1 ? pv[6] : pv[1];
  const float bpw = r1 ? pv[7] : pv[2];
  const float bph = r1 ? pv[8] : pv[3];
  const float bpc = r1 ? pv[9] : pv[4];
  const float btx = r1 ? tv[5] : tv[0];
  const float bty = r1 ? tv[6] : tv[1];
  const float btw = r1 ? tv[7] : tv[2];
  const float bth = r1 ? tv[8] : tv[3];

  const float dx = bpx - btx, dy = bpy - bty;
  const float dw = bpw - btw, dh = bph - bth;
  const float l_xywh = dx * dx + dy * dy + dw * dw + dh * dh;
  const float dcf = bpc - miou;
  const float l_conf = dcf * dcf;

  float l_cls = 0.0f;
#pragma unroll
  for (int d = 10; d < cfg::D; ++d) {
    const float e = pv[d] - tv[d];
    l_cls += e * e;
  }

  const float partial =
      obj * (cfg::LC * l_xywh + l_conf + l_cls) + cfg::LN * noobj * l_noobj;

  // Wave reduction via WMMA (EXEC is all-1s: no divergence reaches here).
  const float wsum = wave_sum32(partial);

  __shared__ float warp_sums[cfg::BLOCK / 32];
  const int lane = threadIdx.x & 31;
  const int wave = threadIdx.x >> 5;
  if (lane == 0) warp_sums[wave] = wsum;
  __syncthreads();

  if (threadIdx.x == 0) {
    float bsum = 0.0f;
#pragma unroll
    for (int w = 0; w < cfg::BLOCK / 32; ++w) bsum += warp_sums[w];
    block_sums[blockIdx.x] = bsum;
  }
}

__global__ __launch_bounds__(cfg::BLOCK)
void yolo_loss_stage2(const float* __restrict__ block_sums,
                      float* __restrict__ out) {
  // Deterministic: fixed strided per-thread accumulation order, then the same
  // WMMA wave reduction and a serial 8-way combine by thread 0.
  float acc = 0.0f;
  for (int i = threadIdx.x; i < cfg::GRID; i += cfg::BLOCK)
    acc += block_sums[i];

  const float wsum = wave_sum32(acc);

  __shared__ float warp_sums[cfg::BLOCK / 32];
  const int lane = threadIdx.x & 31;
  const int wave = threadIdx.x >> 5;
  if (lane == 0) warp_sums[wave] = wsum;
  __syncthreads();

  if (threadIdx.x == 0) {
    float total = 0.0f;
#pragma unroll
    for (int w = 0; w < cfg::BLOCK / 32; ++w) total += warp_sums[w];
    out[0] = total * (1.0f / (float)cfg::NBATCH);
  }
}

extern "C" void kernel_launch(void* const* d_in, const int* in_sizes, int n_in,
                              void* d_out, int out_size, void* d_ws, size_t ws_size,
                              hipStream_t stream) {
  (void)in_sizes; (void)n_in; (void)out_size; (void)ws_size;
  const float* pred = (const float*)d_in[0];
  const float* tgt  = (const float*)d_in[1];
  float* out        = (float*)d_out;
  float* ws         = (float*)d_ws;   // needs cfg::GRID floats (12.25 KB)

  yolo_loss_stage1<<<cfg::GRID, cfg::BLOCK, 0, stream>>>(pred, tgt, ws);
  yolo_loss_stage2<<<1, cfg::BLOCK, 0, stream>>>(ws, out);
}